// graphDenoising_7713761264314
// MI455X (gfx1250) — compile-verified
//
#include <hip/hip_runtime.h>

typedef float v2f __attribute__((ext_vector_type(2)));
typedef float v8f __attribute__((ext_vector_type(8)));

#define BEHAVIOR 4
#define DIM      256
#define NROWS    131072

// One wave per 16-row tile.  D = A x B with
//   A (16x4)  = w[k0..k0+3] replicated in every row  -> lane l holds {w[k0+off], w[k0+off+1]}, off = 2*(l>>4)
//   B (4x16)  = X[n0..n0+15][k0..k0+3]^T             -> lane l holds {X[n0+(l&15)][k0+off], X[..][k0+off+1]}
// Both fragments are a single global_load_b64 per lane.  All D rows are equal
// (A rows identical), so acc[0] in lane n (n<16) is dot(X[n0+n], w).
__global__ __launch_bounds__(256, 2)
void rank1_dot_sigmoid_kernel(const float* __restrict__ user_emb,
                              const float* __restrict__ pos_emb,
                              const float* __restrict__ neg_emb,
                              const float* __restrict__ beh_emb,
                              const float* __restrict__ prompt_emb,
                              float* __restrict__ out)
{
    const int lane = threadIdx.x & 31;
    const int wave = threadIdx.x >> 5;
    const int b    = blockIdx.y;
    const long long n0 = ((long long)blockIdx.x * 8 + wave) * 16;

    // w[b] = beh_emb[b] for b < B-1, else prompt_emb  (rank-1 trick from the reference)
    const float* __restrict__ w = (b < BEHAVIOR - 1) ? (beh_emb + (long long)b * DIM)
                                                     : prompt_emb;

    const int row  = lane & 15;
    const int koff = (lane >> 4) << 1;          // 0 for lanes 0-15, 2 for lanes 16-31

    const long long tileBase = ((long long)b * NROWS + n0) * DIM;
    const float* __restrict__ uRow  = user_emb + tileBase + (long long)row * DIM + koff;
    const float* __restrict__ pRow  = pos_emb  + tileBase + (long long)row * DIM + koff;
    const float* __restrict__ nRow  = neg_emb  + tileBase + (long long)row * DIM + koff;
    const float* __restrict__ wLane = w + koff;

    v8f accU = {};
    v8f accP = {};
    v8f accN = {};

#pragma unroll 8
    for (int k0 = 0; k0 < DIM; k0 += 4) {
        const v2f af = *(const v2f*)(wLane + k0);   // A fragment (w broadcast)
        const v2f bu = *(const v2f*)(uRow  + k0);   // B fragments (16 rows, transposed chunk)
        const v2f bp = *(const v2f*)(pRow  + k0);
        const v2f bn = *(const v2f*)(nRow  + k0);
        // 8 args: (neg_a, A, neg_b, B, c_mod, C, reuse_a, reuse_b)
        accU = __builtin_amdgcn_wmma_f32_16x16x4_f32(false, af, false, bu, (short)0, accU, false, false);
        accP = __builtin_amdgcn_wmma_f32_16x16x4_f32(false, af, false, bp, (short)0, accP, false, false);
        accN = __builtin_amdgcn_wmma_f32_16x16x4_f32(false, af, false, bn, (short)0, accN, false, false);
    }

    const float ud = accU[0];
    const float pd = accP[0];
    const float nd = accN[0];

    if (lane < 16) {
        const long long r     = n0 + lane;
        const long long obase = (long long)b * (2LL * NROWS);
        out[obase + r]         = 1.0f / (1.0f + __expf(-(ud * pd)));   // ge_pos
        out[obase + NROWS + r] = 1.0f / (1.0f + __expf(-(ud * nd)));   // ge_neg
    }
}

extern "C" void kernel_launch(void* const* d_in, const int* in_sizes, int n_in,
                              void* d_out, int out_size, void* d_ws, size_t ws_size,
                              hipStream_t stream)
{
    const float* user_emb   = (const float*)d_in[0];
    const float* pos_emb    = (const float*)d_in[1];
    const float* neg_emb    = (const float*)d_in[2];
    const float* beh_emb    = (const float*)d_in[3];
    const float* prompt_emb = (const float*)d_in[4];
    float* out = (float*)d_out;

    dim3 grid(NROWS / 128, BEHAVIOR, 1);   // 8 waves/block * 16 rows/wave = 128 rows/block
    dim3 block(256, 1, 1);
    hipLaunchKernelGGL(rank1_dot_sigmoid_kernel, grid, block, 0, stream,
                       user_emb, pos_emb, neg_emb, beh_emb, prompt_emb, out);
}